// LocationAndConfidenceLoss_39479339384835
// MI455X (gfx1250) — compile-verified
//
#include <hip/hip_runtime.h>
#include <hip/hip_bf16.h>
#include <stdint.h>

#define BATCH 32
#define NTGT  128
#define VOX   262144
#define BLOCK 1024
#define CHUNK 2
#define NT_TILES (VOX / (BLOCK * CHUNK))   // 128 tiles
#define L1_LO  936                          // min value of (u>>20) for p in [0.001, 0.999)
#define L1_BINS 80                          // bins 936..1015
#define L1_REP 16                           // lane replication factor

// ---------- CDNA5 async global->LDS staging (probe-verified builtin) ----------
#if defined(__has_builtin)
#if __has_builtin(__builtin_amdgcn_global_load_async_to_lds_b32)
#define USE_ASYNC 1
#endif
#endif

#ifdef USE_ASYNC
__device__ __forceinline__ void async_copy_b32(const float* g, float* l) {
  // signature per hipcc diagnostic: (int* gaddr, int* lds, int offset, int cpol)
  __builtin_amdgcn_global_load_async_to_lds_b32(
      (int*)(uintptr_t)g, (int*)l, 0, 0);
}
#if __has_builtin(__builtin_amdgcn_s_wait_asynccnt)
#define WAIT_ASYNC(n) __builtin_amdgcn_s_wait_asynccnt(n)
#else
#define WAIT_ASYNC(n) asm volatile("s_wait_asynccnt %0" :: "i"(n) : "memory")
#endif
#endif

// -(clog(x)) = -max(log(x), -100)
__device__ __forceinline__ float neg_clog(float x) {
  return -fmaxf(logf(x), -100.0f);
}

// Find bin of r-th largest among hist_[0..nbins-1] (bin order = value order).
// Outputs: *s_bin_ = bin index, *s_rem_ = rank within that bin (>=1).
// Grouped scan: 32-bin group sums in parallel, then short serial walks.
__device__ __forceinline__ void find_kth(const uint32_t* hist_, uint32_t* gsum_,
                                         int nbins, int tid, int r,
                                         int* s_bin_, int* s_rem_) {
  const int G = nbins >> 5;
  if (tid < G) {
    uint32_t s = 0;
#pragma unroll
    for (int j = 0; j < 32; ++j) s += hist_[(tid << 5) + j];
    gsum_[tid] = s;
  }
  __syncthreads();
  if (tid == 0) {
    int cum = 0, g = G - 1;
    while (g > 0 && cum + (int)gsum_[g] < r) { cum += (int)gsum_[g]; --g; }
    int bin = (g << 5) + 31;
    while (bin > (g << 5) && cum + (int)hist_[bin] < r) { cum += (int)hist_[bin]; --bin; }
    *s_bin_ = bin;
    *s_rem_ = r - cum;   // rank within bin, in [1, hist_[bin]]
  }
  __syncthreads();
}

__global__ void __launch_bounds__(BLOCK)
loss_main_kernel(const float* __restrict__ pred,      // (B, V, 4)
                 const float* __restrict__ targ,      // (B, N, 3)
                 const float* __restrict__ defs,      // (V, 3)
                 const float* __restrict__ interval,  // (3,)
                 float* __restrict__ partials)        // (B, 2)
{
  __shared__ uint32_t bitmap[VOX / 32];             // 32 KB positive-voxel bitmap
  __shared__ uint32_t histA[L1_BINS * L1_REP];      //  5 KB lane-replicated L1 hist
  __shared__ uint32_t hist[2048];                   //  8 KB L2/L3 hist (aliased as fred)
  __shared__ uint32_t gsum[128];                    //  group sums / L1 totals
  __shared__ float    stage[2 * CHUNK * BLOCK];     // 16 KB async double-buffer
  __shared__ uint32_t s_poscnt;
  __shared__ int      s_bin1, s_bin2, s_bin3, s_rank;
  float* const fred = reinterpret_cast<float*>(hist);   // reduction scratch alias

  const int b   = blockIdx.x;
  const int tid = threadIdx.x;

  for (int i = tid; i < VOX / 32; i += BLOCK) bitmap[i] = 0u;
  for (int i = tid; i < L1_BINS * L1_REP; i += BLOCK) histA[i] = 0u;
  if (tid == 0) s_poscnt = 0u;
  __syncthreads();

  const float iv0 = interval[0], iv1 = interval[1], iv2 = interval[2];
  const int nd0 = (int)(1.0f / iv0);
  const int nd1 = (int)(1.0f / iv1);
  const int nd2 = (int)(1.0f / iv2);

  // ---- targets: bitmap scatter + per-target location loss ----
  float loc_part = 0.0f;
  if (tid < NTGT) {
    const float* t3 = targ + ((size_t)b * NTGT + tid) * 3;
    const float tx = t3[0], ty = t3[1], tz = t3[2];
    const int vx = (int)(tx * (float)nd0);
    const int vy = (int)(ty * (float)nd1);
    const int vz = (int)(tz * (float)nd2);
    const int flat = vx + nd0 * vy + nd0 * nd1 * vz;
    atomicOr(&bitmap[flat >> 5], 1u << (flat & 31));
    const float* pr = pred + (((size_t)b * VOX + (size_t)flat) << 2);
    const float* df = defs + (size_t)flat * 3;
    loc_part = fabsf(pr[0] - (tx - df[0]) / iv0)
             + fabsf(pr[1] - (ty - df[1]) / iv1)
             + fabsf(pr[2] - (tz - df[2]) / iv2);
  }
  __syncthreads();

  // deterministic tree reduction for location loss (fred aliases hist; re-used later)
  fred[tid] = loc_part;
  __syncthreads();
  for (int s = BLOCK / 2; s > 0; s >>= 1) {
    if (tid < s) fred[tid] += fred[tid + s];
    __syncthreads();
  }
  if (tid == 0) partials[2 * b + 0] = fred[0];

  // ---- count unique positives ----
  uint32_t pc = 0;
  for (int i = tid; i < VOX / 32; i += BLOCK) pc += (uint32_t)__popc(bitmap[i]);
  atomicAdd(&s_poscnt, pc);
  __syncthreads();
  const int K = min(3 * (int)s_poscnt, VOX);
  if (K == 0) {               // uniform across block
    if (tid == 0) partials[2 * b + 1] = 0.0f;
    return;
  }

  const float* confp = pred + (size_t)b * VOX * 4 + 3;  // conf channel, stride 4 floats

#ifdef USE_ASYNC
#define CONF_ISSUE(t_)                                                      \
    do {                                                                    \
      const int bb_ = ((t_) & 1) * (BLOCK * CHUNK);                         \
      const int ib_ = (t_) * (BLOCK * CHUNK) + tid;                         \
      _Pragma("unroll")                                                     \
      for (int k_ = 0; k_ < CHUNK; ++k_)                                    \
        async_copy_b32(confp + (size_t)(ib_ + k_ * BLOCK) * 4,              \
                       &stage[bb_ + k_ * BLOCK + tid]);                     \
    } while (0)

#define CONF_STREAM(...)                                                    \
  do {                                                                      \
    CONF_ISSUE(0);                                                          \
    for (int t_ = 0; t_ < NT_TILES; ++t_) {                                 \
      if (t_ + 1 < NT_TILES) { CONF_ISSUE(t_ + 1); WAIT_ASYNC(CHUNK); }     \
      else                   { WAIT_ASYNC(0); }                             \
      const int bb_ = (t_ & 1) * (BLOCK * CHUNK);                           \
      const int base_ = t_ * (BLOCK * CHUNK) + tid;                         \
      _Pragma("unroll")                                                     \
      for (int k_ = 0; k_ < CHUNK; ++k_) {                                  \
        const int i = base_ + k_ * BLOCK;                                   \
        const float pv = stage[bb_ + k_ * BLOCK + tid];                     \
        __VA_ARGS__;                                                        \
      }                                                                     \
    }                                                                       \
  } while (0)
#else
#define CONF_STREAM(...)                                                    \
  do {                                                                      \
    for (int t_ = 0; t_ < NT_TILES; ++t_) {                                 \
      const int base_ = t_ * (BLOCK * CHUNK) + tid;                         \
      _Pragma("unroll")                                                     \
      for (int k_ = 0; k_ < CHUNK; ++k_) {                                  \
        const int i = base_ + k_ * BLOCK;                                   \
        const float pv = confp[(size_t)i * 4];                              \
        __VA_ARGS__;                                                        \
      }                                                                     \
    }                                                                       \
  } while (0)
#endif

  // ---- Pass A: positive BCE sum + level-1 histogram (bits [30:20], replicated) ----
  float pos_sum = 0.0f;
  CONF_STREAM({
    const uint32_t w = bitmap[i >> 5];
    if ((w >> (i & 31)) & 1u) {
      pos_sum += neg_clog(pv);                       // -clog(p) at positives
    } else {
      int bin = (int)(__float_as_uint(pv) >> 20) - L1_LO;
      bin = bin < 0 ? 0 : (bin > L1_BINS - 1 ? L1_BINS - 1 : bin);
      atomicAdd(&histA[bin * L1_REP + (tid & (L1_REP - 1))], 1u);
    }
  });
  __syncthreads();

  // reduce replicated L1 hist, then short serial scan (80 bins)
  if (tid < L1_BINS) {
    uint32_t s = 0;
#pragma unroll
    for (int j = 0; j < L1_REP; ++j) s += histA[tid * L1_REP + j];
    gsum[tid] = s;
  }
  __syncthreads();
  if (tid == 0) {
    int cum = 0, g = L1_BINS - 1;
    while (g > 0 && cum + (int)gsum[g] < K) { cum += (int)gsum[g]; --g; }
    s_bin1 = g + L1_LO;         // absolute bits [30:20] value
    s_rank = K - cum;           // rank within bin1
  }
  __syncthreads();
  const int bin1 = s_bin1;
  for (int i = tid; i < 2048; i += BLOCK) hist[i] = 0u;
  __syncthreads();

  // ---- Pass B: level-2 histogram (bits [19:9]) within bin1 ----
  CONF_STREAM({
    if (!((bitmap[i >> 5] >> (i & 31)) & 1u)) {
      const uint32_t u = __float_as_uint(pv);
      if ((int)(u >> 20) == bin1) atomicAdd(&hist[(u >> 9) & 0x7FFu], 1u);
    }
  });
  __syncthreads();
  find_kth(hist, gsum, 2048, tid, s_rank, &s_bin2, &s_rank);
  const int bin2 = s_bin2;
  const uint32_t key2 = ((uint32_t)bin1 << 11) | (uint32_t)bin2;
  for (int i = tid; i < 2048; i += BLOCK) hist[i] = 0u;
  __syncthreads();

  // ---- Pass C: level-3 histogram (bits [8:0]) -> exact threshold bits ----
  CONF_STREAM({
    if (!((bitmap[i >> 5] >> (i & 31)) & 1u)) {
      const uint32_t u = __float_as_uint(pv);
      if ((u >> 9) == key2) atomicAdd(&hist[u & 0x1FFu], 1u);
    }
  });
  __syncthreads();
  find_kth(hist, gsum, 512, tid, s_rank, &s_bin3, &s_rank);
  const uint32_t tbits = (key2 << 9) | (uint32_t)s_bin3;  // K-th largest p, exact bits
  const int ties = s_rank;                                // # threshold-equal to include
  __syncthreads();   // hist reads done; fred alias reused below

  // ---- Pass D: sum BCE of negatives strictly above threshold ----
  float neg_sum = 0.0f;
  CONF_STREAM({
    if (!((bitmap[i >> 5] >> (i & 31)) & 1u)) {
      if (__float_as_uint(pv) > tbits) neg_sum += neg_clog(1.0f - pv);
    }
  });

  fred[tid] = pos_sum + neg_sum;
  __syncthreads();
  for (int s = BLOCK / 2; s > 0; s >>= 1) {
    if (tid < s) fred[tid] += fred[tid + s];
    __syncthreads();
  }
  if (tid == 0) {
    const float tp = __uint_as_float(tbits);
    partials[2 * b + 1] = fred[0] + (float)ties * neg_clog(1.0f - tp);
  }
}

__global__ void loss_finalize_kernel(const float* __restrict__ partials,
                                     float* __restrict__ out) {
  if (blockIdx.x == 0 && threadIdx.x == 0) {
    float l = 0.0f, c = 0.0f;
    for (int b = 0; b < BATCH; ++b) {   // fixed order -> deterministic
      l += partials[2 * b + 0];
      c += partials[2 * b + 1];
    }
    out[0] = l / (float)BATCH;
    out[1] = c / (float)BATCH;
  }
}

extern "C" void kernel_launch(void* const* d_in, const int* in_sizes, int n_in,
                              void* d_out, int out_size, void* d_ws, size_t ws_size,
                              hipStream_t stream) {
  const float* pred     = (const float*)d_in[0];  // (B, V, 4)
  const float* targ     = (const float*)d_in[1];  // (B, N, 3)
  const float* defs     = (const float*)d_in[2];  // (V, 3)
  const float* interval = (const float*)d_in[3];  // (3,)
  float* partials = (float*)d_ws;                 // B*2 floats
  float* out = (float*)d_out;                     // 2 floats

  loss_main_kernel<<<dim3(BATCH), dim3(BLOCK), 0, stream>>>(
      pred, targ, defs, interval, partials);
  loss_finalize_kernel<<<dim3(1), dim3(1), 0, stream>>>(partials, out);
  (void)in_sizes; (void)n_in; (void)out_size; (void)ws_size;
}